// SparseArch_56745107915216
// MI455X (gfx1250) — compile-verified
//
#include <hip/hip_runtime.h>

typedef __attribute__((ext_vector_type(2))) float v2f;
typedef __attribute__((ext_vector_type(4))) float v4f;
typedef __attribute__((ext_vector_type(8))) float v8f;
typedef __attribute__((ext_vector_type(4))) int   v4i;

// Problem constants (match reference)
constexpr int F = 4;
constexpr int B = 16384;
constexpr int L = 12;
constexpr int V = 1000000;
constexpr int E = 64;

// d_out layout: [0] = loss, [1 .. 1+B*F*E) = pred (row-major [B, F*E])
__global__ void zero_loss_kernel(float* out) {
    if (threadIdx.x == 0) out[0] = 0.0f;
}

__global__ __launch_bounds__(256)
void pos_weighted_bag_kernel(const float* __restrict__ tables,     // [F, V, E]
                             const float* __restrict__ pos_weights,// [F, L]
                             const int*   __restrict__ indices,    // [F, B, L]
                             const int*   __restrict__ lengths,    // [F, B]
                             float* __restrict__ out)              // [1 + B*F*E]
{
    const int lane    = threadIdx.x & 31;   // wave32
    const int wave    = threadIdx.x >> 5;   // 8 waves / block
    const int bagSlot = lane >> 4;          // each wave: 2 bags
    const int lanePos = lane & 15;          // 16 lanes * float4 = 64 floats/row

    // Bag id g in [0, F*B): g = b*F + f so the wave's two bags share b and
    // have adjacent f -> the wave's stores form one contiguous 512B span.
    const int g = ((blockIdx.x * 8 + wave) << 1) + bagSlot;
    const int f = g & (F - 1);
    const int b = g >> 2;

    // ---- bag metadata: 12 indices (48B) and 12 weights (48B) are contiguous
    // and 16B-aligned -> 3+3 b128 loads. All 16 lanes of the half-wave hit the
    // same addresses -> one coalesced request, L1-hot. No DS traffic here.
    const int len = lengths[f * B + b];
    const v4i* ip = (const v4i*)(indices + (size_t)(f * B + b) * L);
    const v4i  i0 = ip[0], i1 = ip[1], i2 = ip[2];
    const v4f* wp = (const v4f*)(pos_weights + f * L);
    const v4f  w0 = wp[0], w1 = wp[1], w2 = wp[2];

    int   idxs[L];
    float ws[L];
#pragma unroll
    for (int i = 0; i < 4; ++i) {
        idxs[i] = i0[i]; idxs[4 + i] = i1[i]; idxs[8 + i] = i2[i];
        ws[i]   = w0[i]; ws[4 + i]   = w1[i]; ws[8 + i]   = w2[i];
    }

    // ---- fully unrolled branchless gather + weighted accumulate.
    // Invalid positions: weight 0, index 0 (row 0 stays hot in cache -> no HBM).
    // Byte offset (f*V+idx)*256 + lanePos*16 < 2^30 fits 32 bits, so express
    // addresses as uniform-base + zext(u32 voffset) to get the saddr global
    // load form (no per-gather 64-bit VALU address math).
    const char* tbase  = (const char*)tables;
    const unsigned fvb = (unsigned)(f * V) << 8;       // feature table byte base
    const unsigned eoff = (unsigned)lanePos << 4;      // lane's 16B slice in row

    v4f acc = {0.0f, 0.0f, 0.0f, 0.0f};
#pragma unroll
    for (int l = 0; l < L; ++l) {
        const bool     valid = l < len;
        const unsigned idx   = valid ? (unsigned)idxs[l] : 0u;
        const float    w     = valid ? ws[l] : 0.0f;
        const unsigned off   = fvb + (idx << 8) + eoff;   // bytes into tables
        const v4f      r     = *(const v4f*)(tbase + off);
        acc.x = fmaf(w, r.x, acc.x);
        acc.y = fmaf(w, r.y, acc.y);
        acc.z = fmaf(w, r.z, acc.z);
        acc.w = fmaf(w, r.w, acc.w);
    }

    // ---- streaming store of pred (nontemporal: don't evict the 1GB table from L2)
    float* dst = out + 1 + (size_t)b * (F * E) + (size_t)f * E + lanePos * 4;
    __builtin_nontemporal_store(acc, (v4f*)dst);

    // ---- loss partial: reduce 32 per-lane sums through the matrix pipe.
    // A(16x4): A[m,0]=lane m sum (lanes 0-15), A[m,2]=lane m+16 sum; A[m,1]=A[m,3]=0.
    // B(4x16)=ones  =>  D[m,n] = s_m + s_{m+16} for every column n.
    const float lane_sum = acc.x + acc.y + acc.z + acc.w;
    v2f a;    a.x = lane_sum; a.y = 0.0f;
    v2f ones; ones.x = 1.0f;  ones.y = 1.0f;
    v8f c = {};
    v8f d = __builtin_amdgcn_wmma_f32_16x16x4_f32(
        /*neg_a=*/false, a, /*neg_b=*/false, ones,
        /*c_mod=*/(short)0, c, /*reuse_a=*/false, /*reuse_b=*/false);
    // Per-lane: sum the 8 D VGPRs (rows 0-7 on lanes 0-15, rows 8-15 on 16-31),
    // then one shfl_xor(16) completes the full 32-lane reduction.
    float t = 0.0f;
#pragma unroll
    for (int i = 0; i < 8; ++i) t += d[i];
    t += __shfl_xor(t, 16, 32);

    __shared__ float wsums[8];
    if (lane == 0) wsums[wave] = t;
    __syncthreads();
    if (threadIdx.x == 0) {
        float s = 0.0f;
#pragma unroll
        for (int i = 0; i < 8; ++i) s += wsums[i];
        // loss = sum(pred) / (B*F*E); 1/4194304 is an exact power of two.
        atomicAdd(out, s * 2.384185791015625e-07f);
    }
}

extern "C" void kernel_launch(void* const* d_in, const int* in_sizes, int n_in,
                              void* d_out, int out_size, void* d_ws, size_t ws_size,
                              hipStream_t stream) {
    const float* tables      = (const float*)d_in[0];
    const float* pos_weights = (const float*)d_in[1];
    const int*   indices     = (const int*)d_in[2];
    const int*   lengths     = (const int*)d_in[3];
    float*       out         = (float*)d_out;

    zero_loss_kernel<<<1, 32, 0, stream>>>(out);
    // F*B = 65536 bags, 16 bags per 256-thread block -> 4096 blocks
    pos_weighted_bag_kernel<<<(F * B) / 16, 256, 0, stream>>>(
        tables, pos_weights, indices, lengths, out);
}